// Decoder_52596169507107
// MI455X (gfx1250) — compile-verified
//
#include <hip/hip_runtime.h>
#include <hip/hip_bf16.h>

typedef __attribute__((ext_vector_type(2))) float v2f;
typedef __attribute__((ext_vector_type(8))) float v8f;

#define BATCH 1024
#define PDIM  32
#define CDIM  224
#define NDIM  441
#define NPAD  448          // padded N stride for LDS (7*64 words -> 2addr_stride64 friendly)
#define PW    33           // padded row stride for staged w (bank-conflict-free halves)
#define NWAVES 8           // 256 threads = 8 wave32
#define NT_TILES 28        // ceil(441/16)
#define CT_TILES 14        // 224/16

__global__ __launch_bounds__(256) void decoder_fused(
    const float* __restrict__ a,      // (B,P,N)
    const float* __restrict__ Yp,     // (B,C,N)
    const float* __restrict__ Wlde,   // (C,P)
    const float* __restrict__ w3c,    // (C,P)  (w3_c[0])
    const float* __restrict__ sigma,  // (1,)
    const float* __restrict__ Wca,    // (2,P)  (W_ca[:,:,0])
    float* __restrict__ y_rec,        // (B,C,N)
    float* __restrict__ y_nl)         // (B,N)
{
    __shared__ float sA[PDIM * NPAD];   // a[b], zero-padded cols 441..447   (57344 B)
    __shared__ float sH[PDIM * NPAD];   // h[b] (P x N)                      (57344 B)
    __shared__ float sW[CDIM * PW];     // w3c staged, padded rows           (29568 B)
    __shared__ float sW2[PDIM];
    __shared__ float sAvg[PDIM];
    __shared__ float sMx[PDIM];
    __shared__ float sHc[PDIM];
    __shared__ float sProp[2];

    const int b     = blockIdx.x;
    const int tid   = threadIdx.x;
    const int lane  = tid & 31;
    const int wave  = tid >> 5;
    const int lhalf = lane & 15;           // M/N index within fragment
    const int khi   = (lane >> 4) << 1;    // +0 / +2 K offset (A,B frags)
    const int phi   = (lane >> 4) << 3;    // +0 / +8 row offset (D frag)

    const float sg = sigma[0];
    const float nh_invs2 = -0.5f / (sg * sg);

    // ---- Phase 0: stage a[b] and w in LDS ----
    for (int i = tid; i < PDIM * NPAD; i += 256) {
        int p = i / NPAD, n = i % NPAD;
        sA[i] = (n < NDIM) ? a[((size_t)b * PDIM + p) * NDIM + n] : 0.f;
    }
    for (int i = tid; i < CDIM * PDIM; i += 256) {
        int c = i >> 5, p = i & 31;
        sW[c * PW + p] = w3c[i];
    }
    __syncthreads();
    if (tid < PDIM) {                       // w2[p] = sum_c w[c][p]^2
        float s = 0.f;
        for (int c = 0; c < CDIM; ++c) { float v = sW[c * PW + tid]; s += v * v; }
        sW2[tid] = s;
    }
    __syncthreads();

    // ---- Phase 1: y_rec[b] = W_lde (224x32) @ a[b] (32x441), WMMA f32 16x16x4 ----
    for (int t = wave; t < CT_TILES * NT_TILES; t += NWAVES) {
        int ct = t / NT_TILES, nt = t % NT_TILES;
        int cbase = ct * 16, nbase = nt * 16;
        v8f acc = {};
        #pragma unroll
        for (int k0 = 0; k0 < PDIM; k0 += 4) {
            int ka = k0 + khi;
            v2f af, bf;
            af.x = Wlde[(cbase + lhalf) * PDIM + ka];
            af.y = Wlde[(cbase + lhalf) * PDIM + ka + 1];
            bf.x = sA[ka * NPAD + nbase + lhalf];
            bf.y = sA[(ka + 1) * NPAD + nbase + lhalf];
            acc = __builtin_amdgcn_wmma_f32_16x16x4_f32(
                false, af, false, bf, (short)0, acc, false, false);
        }
        int n = nbase + lhalf;
        if (n < NDIM) {
            size_t base = ((size_t)b * CDIM + cbase + phi) * NDIM + n;
            #pragma unroll
            for (int r = 0; r < 8; ++r)
                y_rec[base + (size_t)r * NDIM] = acc[r];
        }
    }

    // ---- Phase 2: xw^T = w^T(32x224) @ Y_patch[b](224x441); fused x2 and h ----
    // Each wave owns an n-tile and computes both P-tiles, so Y_patch streams once.
    for (int nt = wave; nt < NT_TILES; nt += NWAVES) {
        int nbase = nt * 16;
        int n = nbase + lhalf;
        bool nv = (n < NDIM);
        int nc = nv ? n : (NDIM - 1);           // clamped address
        float msk = nv ? 1.f : 0.f;             // zero out-of-range columns
        const float* Yb = Yp + (size_t)b * CDIM * NDIM;
        v8f acc0 = {}, acc1 = {};
        float s2 = 0.f;
        #pragma unroll 2
        for (int k0 = 0; k0 < CDIM; k0 += 4) {
            int ka = k0 + khi;
            float b0 = Yb[(size_t)ka * NDIM + nc] * msk;
            float b1 = Yb[(size_t)(ka + 1) * NDIM + nc] * msk;
            s2 += b0 * b0 + b1 * b1;
            v2f bf; bf.x = b0; bf.y = b1;
            v2f a0, a1;                          // A = w^T : A[p][c] = w[c][p]
            a0.x = sW[ka * PW + lhalf];
            a0.y = sW[(ka + 1) * PW + lhalf];
            a1.x = sW[ka * PW + 16 + lhalf];
            a1.y = sW[(ka + 1) * PW + 16 + lhalf];
            acc0 = __builtin_amdgcn_wmma_f32_16x16x4_f32(
                false, a0, false, bf, (short)0, acc0, false, false);
            acc1 = __builtin_amdgcn_wmma_f32_16x16x4_f32(
                false, a1, false, bf, (short)0, acc1, false, false);
        }
        // lanes L and L^16 hold the two K-halves of column n's sum of squares
        float x2 = s2 + __shfl_xor(s2, 16, 32);
        if (nv) {
            #pragma unroll
            for (int r = 0; r < 8; ++r) {
                int p0 = phi + r;
                float sq0 = fmaxf(x2 + sW2[p0] - 2.f * acc0[r], 0.f);
                sH[p0 * NPAD + n] = expf(nh_invs2 * sq0);
                int p1 = 16 + p0;
                float sq1 = fmaxf(x2 + sW2[p1] - 2.f * acc1[r], 0.f);
                sH[p1 * NPAD + n] = expf(nh_invs2 * sq1);
            }
        }
    }
    __syncthreads();

    // ---- Phase 3: per-row avg & max of h (8 lanes per row + shfl tree) ----
    {
        int p = tid >> 3;                  // 0..31
        int j = tid & 7;
        float s = 0.f, m = -3.402823466e38f;
        for (int n = j; n < NDIM; n += 8) {
            float v = sH[p * NPAD + n];
            s += v; m = fmaxf(m, v);
        }
        #pragma unroll
        for (int o = 1; o < 8; o <<= 1) {
            s += __shfl_xor(s, o, 32);
            m = fmaxf(m, __shfl_xor(m, o, 32));
        }
        if (j == 0) { sAvg[p] = s * (1.f / (float)NDIM); sMx[p] = m; }
    }
    __syncthreads();

    // ---- Phase 4: gate: leaky_relu matvecs, softmax(2), hc ----
    if (tid == 0) {
        float y0a = 0.f, y0m = 0.f, y1a = 0.f, y1m = 0.f;
        for (int p = 0; p < PDIM; ++p) {
            float wc0 = Wca[p], wc1 = Wca[PDIM + p];
            y0a += sAvg[p] * wc0; y0m += sMx[p] * wc0;
            y1a += sAvg[p] * wc1; y1m += sMx[p] * wc1;
        }
        float y0 = (y0a > 0.f ? y0a : 0.01f * y0a) + (y0m > 0.f ? y0m : 0.01f * y0m);
        float y1 = (y1a > 0.f ? y1a : 0.01f * y1a) + (y1m > 0.f ? y1m : 0.01f * y1m);
        float mxy = fmaxf(y0, y1);
        float e0 = expf(y0 - mxy), e1 = expf(y1 - mxy);
        float inv = 1.f / (e0 + e1);
        sProp[0] = e0 * inv; sProp[1] = e1 * inv;
    }
    __syncthreads();
    if (tid < PDIM)
        sHc[tid] = sProp[0] * sH[tid * NPAD + (NDIM / 2)] + sProp[1] * sAvg[tid];
    __syncthreads();

    // ---- Phase 5: y_nl_rec[b][n] = sum_p hc[p] * a[b][p][n] ----
    for (int n = tid; n < NDIM; n += 256) {
        float s = 0.f;
        #pragma unroll
        for (int p = 0; p < PDIM; ++p) s += sHc[p] * sA[p * NPAD + n];
        y_nl[(size_t)b * NDIM + n] = s;
    }
}

extern "C" void kernel_launch(void* const* d_in, const int* in_sizes, int n_in,
                              void* d_out, int out_size, void* d_ws, size_t ws_size,
                              hipStream_t stream) {
    (void)in_sizes; (void)n_in; (void)out_size; (void)d_ws; (void)ws_size;
    const float* a     = (const float*)d_in[0];
    const float* Yp    = (const float*)d_in[1];
    const float* Wlde  = (const float*)d_in[2];
    const float* w3c   = (const float*)d_in[3];   // (1,C,P) -> flat (C,P)
    const float* sigma = (const float*)d_in[4];
    const float* Wca   = (const float*)d_in[5];   // (2,P,1) -> flat (2,P)
    float* y_rec = (float*)d_out;
    float* y_nl  = y_rec + (size_t)BATCH * CDIM * NDIM;

    decoder_fused<<<dim3(BATCH), dim3(256), 0, stream>>>(
        a, Yp, Wlde, w3c, sigma, Wca, y_rec, y_nl);
}